// ImprovedAudioDDCMCodebook_2044404433531
// MI455X (gfx1250) — compile-verified
//
#include <hip/hip_runtime.h>

typedef __attribute__((ext_vector_type(2))) float v2f;
typedef __attribute__((ext_vector_type(8))) float v8f;

#define KDIM   32000
#define BATCH  128
#define CODES  1024
#define KCHUNK 1000
#define NKCH   32

// workspace layout (float offsets)
#define WS_DOT    0           // 128*1024
#define WS_LNORM  131072      // 128
#define WS_CNORM  131200      // 1024
#define WS_W      132224      // 128*5 weights
#define WS_IDX    132864      // 128*5 ints
#define WS_ZERON  132224      // floats to zero

// output layout (float offsets): quantized, main_idx, main_dist, usage_new
#define OUT_IDX   4096000
#define OUT_DIST  4096128
#define OUT_USAGE 4096256

__global__ void __launch_bounds__(256) vq_init_kernel(float* __restrict__ ws,
                                                      float* __restrict__ out,
                                                      const float* __restrict__ usage) {
  const int i = blockIdx.x * 256 + threadIdx.x;
  if (i < WS_ZERON) ws[i] = 0.0f;
  if (i < CODES) out[OUT_USAGE + i] = usage[i];
}

// 128x1024 GEMM (dot = lf @ cf.T), K=32000, fp32 WMMA 16x16x4.
// Block = 8 waves sharing one 16x64 C-slab (mtile, ngroup); wave w owns
// K-chunk (kcg*8 + w). Inner loop: 8 K per trip (8 WMMAs, 10 b64 loads)
// with constant-offset B-tile addressing (24-bit immediates). The two
// K-step halves use independent accumulator sets so the allocator never
// has to shuffle accumulator VGPRs between WMMA groups.
// Partials reduce through LDS (ds_add_f32); wave 0 flushes one set of
// global atomics per block. Row norms are folded into the same pass.
__global__ void __launch_bounds__(256) vq_gemm_kernel(const float* __restrict__ lf,
                                                      const float* __restrict__ cf,
                                                      float* __restrict__ ws) {
  __shared__ float slab[16 * 64];   // C-slab accumulator, 4 KB

  float* dot   = ws + WS_DOT;
  float* lnorm = ws + WS_LNORM;
  float* cnorm = ws + WS_CNORM;

  const int kcg    = blockIdx.x >> 7;        // 0..3  (group of 8 K-chunks)
  const int rem    = blockIdx.x & 127;
  const int mtile  = rem >> 4;               // 0..7
  const int ngroup = rem & 15;               // 0..15
  const int wave   = threadIdx.x >> 5;       // 0..7
  const int kchunk = kcg * 8 + wave;         // 0..31
  const int lane   = threadIdx.x & 31;
  const int l16    = lane & 15;
  const int hlf    = lane >> 4;              // K phase: 0 -> K0/K1, 1 -> K2/K3

  // zero the LDS slab (coalesced, 4 floats/thread)
  slab[threadIdx.x]       = 0.0f;
  slab[threadIdx.x + 256] = 0.0f;
  slab[threadIdx.x + 512] = 0.0f;
  slab[threadIdx.x + 768] = 0.0f;
  __syncthreads();

  const int brow = mtile * 16 + l16;
  const float* ap = lf + (size_t)brow * KDIM + hlf * 2 + kchunk * KCHUNK;
  const float* bp = cf + (size_t)(ngroup * 64 + l16) * KDIM + hlf * 2 + kchunk * KCHUNK;

  v8f accA0 = {}, accA1 = {}, accA2 = {}, accA3 = {};
  v8f accB0 = {}, accB1 = {}, accB2 = {}, accB3 = {};
  float an = 0.0f, bn0 = 0.0f, bn1 = 0.0f, bn2 = 0.0f, bn3 = 0.0f;

  for (int it = 0; it < KCHUNK / 8; ++it) {     // 125 iterations, 8 K each
    const float2 a0  = *(const float2*)(ap);
    const float2 a1  = *(const float2*)(ap + 4);
    const float2 b00 = *(const float2*)(bp);
    const float2 b01 = *(const float2*)(bp + 4);
    const float2 b10 = *(const float2*)(bp + 16 * KDIM);
    const float2 b11 = *(const float2*)(bp + 16 * KDIM + 4);
    const float2 b20 = *(const float2*)(bp + 32 * KDIM);
    const float2 b21 = *(const float2*)(bp + 32 * KDIM + 4);
    const float2 b30 = *(const float2*)(bp + 48 * KDIM);
    const float2 b31 = *(const float2*)(bp + 48 * KDIM + 4);

    v2f av0 = {a0.x, a0.y},   av1 = {a1.x, a1.y};
    v2f bv00 = {b00.x, b00.y}, bv01 = {b01.x, b01.y};
    v2f bv10 = {b10.x, b10.y}, bv11 = {b11.x, b11.y};
    v2f bv20 = {b20.x, b20.y}, bv21 = {b21.x, b21.y};
    v2f bv30 = {b30.x, b30.y}, bv31 = {b31.x, b31.y};

    accA0 = __builtin_amdgcn_wmma_f32_16x16x4_f32(false, av0, false, bv00, (short)0, accA0, false, false);
    accA1 = __builtin_amdgcn_wmma_f32_16x16x4_f32(false, av0, false, bv10, (short)0, accA1, false, false);
    accA2 = __builtin_amdgcn_wmma_f32_16x16x4_f32(false, av0, false, bv20, (short)0, accA2, false, false);
    accA3 = __builtin_amdgcn_wmma_f32_16x16x4_f32(false, av0, false, bv30, (short)0, accA3, false, false);
    accB0 = __builtin_amdgcn_wmma_f32_16x16x4_f32(false, av1, false, bv01, (short)0, accB0, false, false);
    accB1 = __builtin_amdgcn_wmma_f32_16x16x4_f32(false, av1, false, bv11, (short)0, accB1, false, false);
    accB2 = __builtin_amdgcn_wmma_f32_16x16x4_f32(false, av1, false, bv21, (short)0, accB2, false, false);
    accB3 = __builtin_amdgcn_wmma_f32_16x16x4_f32(false, av1, false, bv31, (short)0, accB3, false, false);

    an  = fmaf(a0.x,  a0.x,  fmaf(a0.y,  a0.y,  an));
    an  = fmaf(a1.x,  a1.x,  fmaf(a1.y,  a1.y,  an));
    bn0 = fmaf(b00.x, b00.x, fmaf(b00.y, b00.y, bn0));
    bn0 = fmaf(b01.x, b01.x, fmaf(b01.y, b01.y, bn0));
    bn1 = fmaf(b10.x, b10.x, fmaf(b10.y, b10.y, bn1));
    bn1 = fmaf(b11.x, b11.x, fmaf(b11.y, b11.y, bn1));
    bn2 = fmaf(b20.x, b20.x, fmaf(b20.y, b20.y, bn2));
    bn2 = fmaf(b21.x, b21.x, fmaf(b21.y, b21.y, bn2));
    bn3 = fmaf(b30.x, b30.x, fmaf(b30.y, b30.y, bn3));
    bn3 = fmaf(b31.x, b31.x, fmaf(b31.y, b31.y, bn3));

    ap += 8;
    bp += 8;
  }

  // Merge the two accumulator sets (32 v_add_f32, once per wave).
  const v8f acc0 = accA0 + accB0;
  const v8f acc1 = accA1 + accB1;
  const v8f acc2 = accA2 + accB2;
  const v8f acc3 = accA3 + accB3;

  // Norm partials: fold the two K phases, then one global atomic per row/col.
  an  += __shfl_xor(an, 16, 32);
  bn0 += __shfl_xor(bn0, 16, 32);
  bn1 += __shfl_xor(bn1, 16, 32);
  bn2 += __shfl_xor(bn2, 16, 32);
  bn3 += __shfl_xor(bn3, 16, 32);
  if (ngroup == 0 && hlf == 0) unsafeAtomicAdd(&lnorm[brow], an);
  if (mtile == 0 && hlf == 0) {
    unsafeAtomicAdd(&cnorm[ngroup * 64 +  0 + l16], bn0);
    unsafeAtomicAdd(&cnorm[ngroup * 64 + 16 + l16], bn1);
    unsafeAtomicAdd(&cnorm[ngroup * 64 + 32 + l16], bn2);
    unsafeAtomicAdd(&cnorm[ngroup * 64 + 48 + l16], bn3);
  }

  // C/D layout: VGPR j, lanes 0-15 -> M=j, lanes 16-31 -> M=j+8, N=lane%16.
  // Accumulate all 8 waves' partials into the LDS slab (ds_add_f32).
#pragma unroll
  for (int j = 0; j < 8; ++j) {
    const int row = hlf * 8 + j;      // 0..15 within slab
    float* srow = slab + row * 64;
    atomicAdd(srow +  0 + l16, acc0[j]);
    atomicAdd(srow + 16 + l16, acc1[j]);
    atomicAdd(srow + 32 + l16, acc2[j]);
    atomicAdd(srow + 48 + l16, acc3[j]);
  }
  __syncthreads();

  // Wave 0 flushes the slab: coalesced 128B global atomic bursts.
  if (wave == 0) {
    float* dbase = dot + (size_t)(mtile * 16) * CODES + ngroup * 64;
#pragma unroll
    for (int j = 0; j < 32; ++j) {
      const int idx = j * 32 + lane;  // 0..1023, lane-contiguous
      const int row = idx >> 6;
      const int col = idx & 63;
      unsafeAtomicAdd(dbase + (size_t)row * CODES + col, slab[idx]);
    }
  }
}

// One wave per latent: distances, top-5 (smallest), softmax, usage scatter.
__global__ void __launch_bounds__(32) vq_topk_kernel(float* __restrict__ ws,
                                                     float* __restrict__ out) {
  const float* dot   = ws + WS_DOT;
  const float* lnorm = ws + WS_LNORM;
  const float* cnorm = ws + WS_CNORM;
  float* wsw = ws + WS_W;
  int*   wsi = (int*)(ws + WS_IDX);

  const int b = blockIdx.x;
  const int lane = threadIdx.x;
  const float ln = lnorm[b];

  float dv[32];
#pragma unroll
  for (int j = 0; j < 32; ++j) {
    const int c = j * 32 + lane;
    const float d2 = ln + cnorm[c] - 2.0f * dot[(size_t)b * CODES + c];
    dv[j] = sqrtf(fmaxf(d2, 0.0f));
  }

  int picked[5] = {-1, -1, -1, -1, -1};
  float pdist[5];
#pragma unroll
  for (int sel = 0; sel < 5; ++sel) {
    float best = 3.4e38f;
    int bc = 0x7fffffff;
#pragma unroll
    for (int j = 0; j < 32; ++j) {
      const int c = j * 32 + lane;
      const bool skip = (c == picked[0]) | (c == picked[1]) | (c == picked[2]) |
                        (c == picked[3]) | (c == picked[4]);
      if (!skip && (dv[j] < best || (dv[j] == best && c < bc))) { best = dv[j]; bc = c; }
    }
#pragma unroll
    for (int off = 16; off >= 1; off >>= 1) {
      const float od = __shfl_xor(best, off, 32);
      const int   oc = __shfl_xor(bc, off, 32);
      if (od < best || (od == best && oc < bc)) { best = od; bc = oc; }
    }
    picked[sel] = bc;
    pdist[sel] = best;
  }

  // softmax(-d/0.1), stabilized around the smallest distance pdist[0]
  float w[5];
  float s = 0.0f;
#pragma unroll
  for (int k = 0; k < 5; ++k) { w[k] = __expf((pdist[0] - pdist[k]) * 10.0f); s += w[k]; }
  const float inv = 1.0f / s;

  if (lane == 0) {
    out[OUT_IDX + b]  = (float)picked[0];
    out[OUT_DIST + b] = pdist[0];
#pragma unroll
    for (int k = 0; k < 5; ++k) {
      const float wk = w[k] * inv;
      wsw[b * 5 + k] = wk;
      wsi[b * 5 + k] = picked[k];
      unsafeAtomicAdd(&out[OUT_USAGE + picked[k]], wk);
    }
  }
}

// quantized[b,:] = sum_k w[b,k] * cf[idx[b,k],:]  (codebook is L2-resident)
__global__ void __launch_bounds__(256) vq_quant_kernel(const float* __restrict__ cf,
                                                       const float* __restrict__ ws,
                                                       float* __restrict__ out) {
  const int b = blockIdx.y;
  const int t = blockIdx.x * 256 + threadIdx.x;
  if (t >= KDIM / 4) return;
  const float* wsw = ws + WS_W + b * 5;
  const int*   wsi = (const int*)(ws + WS_IDX) + b * 5;
  const float w0 = wsw[0], w1 = wsw[1], w2 = wsw[2], w3 = wsw[3], w4 = wsw[4];
  const int   i0 = wsi[0], i1 = wsi[1], i2 = wsi[2], i3 = wsi[3], i4 = wsi[4];
  const int base = t * 4;
  const float4 v0 = *(const float4*)(cf + (size_t)i0 * KDIM + base);
  const float4 v1 = *(const float4*)(cf + (size_t)i1 * KDIM + base);
  const float4 v2 = *(const float4*)(cf + (size_t)i2 * KDIM + base);
  const float4 v3 = *(const float4*)(cf + (size_t)i3 * KDIM + base);
  const float4 v4 = *(const float4*)(cf + (size_t)i4 * KDIM + base);
  float4 r;
  r.x = w0 * v0.x + w1 * v1.x + w2 * v2.x + w3 * v3.x + w4 * v4.x;
  r.y = w0 * v0.y + w1 * v1.y + w2 * v2.y + w3 * v3.y + w4 * v4.y;
  r.z = w0 * v0.z + w1 * v1.z + w2 * v2.z + w3 * v3.z + w4 * v4.z;
  r.w = w0 * v0.w + w1 * v1.w + w2 * v2.w + w3 * v3.w + w4 * v4.w;
  *(float4*)(out + (size_t)b * KDIM + base) = r;
}

extern "C" void kernel_launch(void* const* d_in, const int* in_sizes, int n_in,
                              void* d_out, int out_size, void* d_ws, size_t ws_size,
                              hipStream_t stream) {
  const float* lf    = (const float*)d_in[0];  // latent  (128, 8, 250, 16)
  const float* cf    = (const float*)d_in[1];  // codebook (1024, 8, 250, 16)
  const float* usage = (const float*)d_in[2];  // usage_count (1024,)
  float* out = (float*)d_out;
  float* ws  = (float*)d_ws;

  vq_init_kernel<<<(WS_ZERON + 255) / 256, 256, 0, stream>>>(ws, out, usage);
  vq_gemm_kernel<<<512, 256, 0, stream>>>(lf, cf, ws);           // 4096 waves
  vq_topk_kernel<<<BATCH, 32, 0, stream>>>(ws, out);
  vq_quant_kernel<<<dim3(32, BATCH), 256, 0, stream>>>(cf, ws, out);
}